// SelfAttentionLayer_26087631356005
// MI455X (gfx1250) — compile-verified
//
#include <hip/hip_runtime.h>

typedef float v2f __attribute__((ext_vector_type(2)));
typedef float v8f __attribute__((ext_vector_type(8)));

#define B_SZ  2
#define T_SZ  2048
#define D_SZ  1024
#define H_SZ  16
#define HD_SZ 64
#define ATT_SCALE 0.125f   // 1/sqrt(64)
#define NEG_INF  (-1e9f)

// ---------------------------------------------------------------------------
// CDNA5 async LDS<->global copies (16B per lane, ASYNCcnt-tracked) + wait.
// ---------------------------------------------------------------------------
__device__ __forceinline__ void async_b128(unsigned lds_off, const void* g)
{
    asm volatile("global_load_async_to_lds_b128 %0, %1, off"
                 :: "v"(lds_off), "v"(g) : "memory");
}
__device__ __forceinline__ void async_store_b128(void* g, unsigned lds_off)
{
    asm volatile("global_store_async_from_lds_b128 %0, %1, off"
                 :: "v"(g), "v"(lds_off) : "memory");
}
__device__ __forceinline__ void wait_async_all()
{
    asm volatile("s_wait_asynccnt 0x0" ::: "memory");
}

__device__ __forceinline__ v2f ld2(const float* p)
{
    v2f r; r[0] = p[0]; r[1] = p[1]; return r;
}

// ---------------------------------------------------------------------------
// f32 WMMA GEMM:  C = A[M,K] @ W[K,N] (+ bias)  (used for the 4 projections)
//   128x64 workgroup tile, 8 waves (4x2), each 2x2 16x16 accumulators.
//   A tile: async global->LDS b128, double buffered.
//   W tile: register-staged, stored TRANSPOSED (Ws[n][k]) -> B frag = 1x b64.
//   MODE 0: C row-major [M,N]
//   MODE 1: head-split:  row=b*T+t, col=h*hd+d -> C[((b*H+h)*T+t)*hd+d]
// ---------------------------------------------------------------------------
template<int MODE>
__global__ __launch_bounds__(256) void gemm_f32_wmma(
    const float* __restrict__ A, const float* __restrict__ W,
    const float* __restrict__ bias, float* __restrict__ Cbase,
    int M, int N, int K)
{
    __shared__ float As[2][128][36];   // [buf][m][k], +4 pad (rows 16B aligned)
    __shared__ float Ws[2][64][36];    // [buf][n][k], transposed

    const int tid  = threadIdx.x;
    const int wave = tid >> 5;
    const int lane = tid & 31;
    const int half = lane >> 4;
    const int ml   = lane & 15;
    const int wm   = wave >> 1;        // 0..3
    const int wn   = wave & 1;         // 0..1

    const int m0 = blockIdx.y * 128;
    const int n0 = blockIdx.x * 64;

    const unsigned asBase = (unsigned)(size_t)&As[0][0][0];

    const int arc = (tid & 7) << 2;    // k offset (0,4,..,28)
    const int ar0 = tid >> 3;          // row base (0..31), +32/round
    auto issueA = [&](int k0, int buf) {
        #pragma unroll
        for (int q = 0; q < 4; ++q) {
            const int r = ar0 + q * 32;
            async_b128(asBase + (unsigned)((((buf * 128 + r) * 36) + arc) << 2),
                       A + (long long)(m0 + r) * K + k0 + arc);
        }
    };

    const int wr0 = tid >> 4;          // k row (0..15), +16/round
    const int wc  = (tid & 15) << 2;   // n offset
    float4 wreg[2];
    auto loadW = [&](int k0) {
        #pragma unroll
        for (int q = 0; q < 2; ++q)
            wreg[q] = *(const float4*)(W + (long long)(k0 + wr0 + q * 16) * N + n0 + wc);
    };
    auto storeW = [&](int buf) {
        #pragma unroll
        for (int q = 0; q < 2; ++q) {
            const int r = wr0 + q * 16;
            Ws[buf][wc + 0][r] = wreg[q].x;
            Ws[buf][wc + 1][r] = wreg[q].y;
            Ws[buf][wc + 2][r] = wreg[q].z;
            Ws[buf][wc + 3][r] = wreg[q].w;
        }
    };

    v8f acc[2][2];
    acc[0][0] = {}; acc[0][1] = {}; acc[1][0] = {}; acc[1][1] = {};

    issueA(0, 0);
    loadW(0);

    const int niter = K >> 5;
    for (int it = 0; it < niter; ++it) {
        const int buf = it & 1;
        wait_async_all();
        storeW(buf);
        __syncthreads();
        if (it + 1 < niter) {
            issueA((it + 1) << 5, buf ^ 1);
            loadW((it + 1) << 5);
        }
        #pragma unroll
        for (int kk = 0; kk < 32; kk += 4) {
            const int ka = kk + 2 * half;
            v2f a[2], b[2];
            #pragma unroll
            for (int i = 0; i < 2; ++i)
                a[i] = ld2(&As[buf][wm * 32 + i * 16 + ml][ka]);
            #pragma unroll
            for (int j = 0; j < 2; ++j)
                b[j] = ld2(&Ws[buf][wn * 32 + j * 16 + ml][ka]);
            #pragma unroll
            for (int i = 0; i < 2; ++i)
                #pragma unroll
                for (int j = 0; j < 2; ++j)
                    acc[i][j] = __builtin_amdgcn_wmma_f32_16x16x4_f32(
                        false, a[i], false, b[j], (short)0, acc[i][j], false, false);
        }
    }

    #pragma unroll
    for (int i = 0; i < 2; ++i) {
        #pragma unroll
        for (int j = 0; j < 2; ++j) {
            const int col = n0 + wn * 32 + j * 16 + ml;
            const float bv = bias ? bias[col] : 0.0f;
            #pragma unroll
            for (int r = 0; r < 8; ++r) {
                const int row = m0 + wm * 32 + i * 16 + 8 * half + r;
                const float val = acc[i][j][r] + bv;
                if (MODE == 0) {
                    Cbase[(long long)row * N + col] = val;
                } else {
                    const int bb = row / T_SZ, t = row % T_SZ;
                    const int h  = col / HD_SZ, d = col % HD_SZ;
                    Cbase[(((long long)bb * H_SZ + h) * T_SZ + t) * HD_SZ + d] = val;
                }
            }
        }
    }
}

// ---------------------------------------------------------------------------
// Fused attention: scores -> softmax -> (att store || P@V), one q-block of 32
// rows per workgroup, probability block resident in 320KB CDNA5 LDS.
//   grid: (T/32, B*H), 256 threads (8 waves: 2 m-tiles x 4 n-tiles).
// ---------------------------------------------------------------------------
__global__ __launch_bounds__(256) void attn_fused(
    const float* __restrict__ Q, const float* __restrict__ Km,
    const float* __restrict__ Vm, const unsigned char* __restrict__ mask,
    float* __restrict__ att, float* __restrict__ Ctx)
{
    __shared__ float P[32][2052];      // 32 q-rows x 2048 keys (+4 pad)
    __shared__ float Qs[32][68];       // q block, row-major [m][k]
    __shared__ float KV[2][64][68];    // K tiles [key][k] / V tiles [n][k]

    const int tid  = threadIdx.x;
    const int wave = tid >> 5;
    const int lane = tid & 31;
    const int half = lane >> 4;
    const int ml   = lane & 15;
    const int wqm  = wave >> 2;        // 0..1  (16-row m-tile)
    const int wqn  = wave & 3;         // 0..3  (16-col n-tile)

    const int z  = blockIdx.y;         // b*H + h
    const int bb = z / H_SZ;
    const int hh = z % H_SZ;
    const int q0 = blockIdx.x * 32;

    const float* Qb = Q  + (long long)z * T_SZ * HD_SZ;
    const float* Kb = Km + (long long)z * T_SZ * HD_SZ;
    const float* Vb = Vm + (long long)z * T_SZ * HD_SZ;
    float* attb = att + (long long)z * T_SZ * T_SZ;

    const unsigned pBase  = (unsigned)(size_t)&P[0][0];
    const unsigned qsBase = (unsigned)(size_t)&Qs[0][0];
    const unsigned kvBase = (unsigned)(size_t)&KV[0][0][0];

    const int c4 = (tid & 15) << 2;    // 0,4,..,60
    const int r0 = tid >> 4;           // 0..15

    // stage Q block (32x64): 512 chunks, 2/thread
    #pragma unroll
    for (int q = 0; q < 2; ++q) {
        const int r = r0 + q * 16;
        async_b128(qsBase + (unsigned)(((r * 68) + c4) << 2),
                   Qb + (long long)(q0 + r) * HD_SZ + c4);
    }
    auto issueK = [&](int kt0, int buf) {
        #pragma unroll
        for (int q = 0; q < 4; ++q) {
            const int r = r0 + q * 16;
            async_b128(kvBase + (unsigned)(((((buf * 64) + r) * 68) + c4) << 2),
                       Kb + (long long)(kt0 + r) * HD_SZ + c4);
        }
    };
    issueK(0, 0);

    // ---- phase 1: masked, scaled scores into LDS P ----
    const int ktiles = T_SZ / 64;      // 32
    for (int it = 0; it < ktiles; ++it) {
        const int buf = it & 1;
        const int kt0 = it * 64;
        wait_async_all();              // tile `it` (and Q on it==0) resident
        __syncthreads();
        if (it + 1 < ktiles) issueK((it + 1) * 64, buf ^ 1);

        v8f s = {};
        #pragma unroll
        for (int kk = 0; kk < HD_SZ; kk += 4) {
            const int ka = kk + 2 * half;
            v2f a = ld2(&Qs[wqm * 16 + ml][ka]);
            v2f b = ld2(&KV[buf][wqn * 16 + ml][ka]);   // B = K^T
            s = __builtin_amdgcn_wmma_f32_16x16x4_f32(
                    false, a, false, b, (short)0, s, false, false);
        }
        const int ktc = kt0 + wqn * 16 + ml;
        const bool masked = mask[bb * T_SZ + ktc] != 0;
        #pragma unroll
        for (int r = 0; r < 8; ++r) {
            const int row = wqm * 16 + 8 * half + r;
            P[row][ktc] = masked ? NEG_INF : s[r] * ATT_SCALE;
        }
    }
    __syncthreads();

    // ---- phase 2: softmax rows in LDS (wave handles 4 rows) ----
    #pragma unroll 1
    for (int rr = 0; rr < 4; ++rr) {
        const int row = wave * 4 + rr;
        float lmax = -3.4e38f;
        for (int i = lane; i < T_SZ; i += 32)
            lmax = fmaxf(lmax, P[row][i]);
        #pragma unroll
        for (int mk = 16; mk > 0; mk >>= 1)
            lmax = fmaxf(lmax, __shfl_xor(lmax, mk, 32));
        float lsum = 0.0f;
        for (int i = lane; i < T_SZ; i += 32) {
            const float e = __expf(P[row][i] - lmax);
            P[row][i] = e;
            lsum += e;
        }
        #pragma unroll
        for (int mk = 16; mk > 0; mk >>= 1)
            lsum += __shfl_xor(lsum, mk, 32);
        const float inv = 1.0f / lsum;
        for (int i = lane; i < T_SZ; i += 32)
            P[row][i] *= inv;
    }
    __syncthreads();

    // ---- phase 3: async store P -> att (drains during P@V below) ----
    #pragma unroll 1
    for (int i = 0; i < 64; ++i) {
        const int c   = tid + (i << 8);
        const int row = c >> 9;              // 512 chunks per row
        const int col = (c & 511) << 2;
        async_store_b128(attb + (long long)(q0 + row) * T_SZ + col,
                         pBase + (unsigned)(((row * 2052) + col) << 2));
    }

    // ---- phase 4: ctx = P @ V (P already in LDS; V register-transposed) ----
    float4 vreg[4];
    auto loadV = [&](int kt0) {
        #pragma unroll
        for (int q = 0; q < 4; ++q)
            vreg[q] = *(const float4*)(Vb + (long long)(kt0 + r0 + q * 16) * HD_SZ + c4);
    };
    auto storeV = [&](int buf) {
        #pragma unroll
        for (int q = 0; q < 4; ++q) {
            const int r = r0 + q * 16;
            KV[buf][c4 + 0][r] = vreg[q].x;
            KV[buf][c4 + 1][r] = vreg[q].y;
            KV[buf][c4 + 2][r] = vreg[q].z;
            KV[buf][c4 + 3][r] = vreg[q].w;
        }
    };
    loadV(0);
    v8f acc = {};
    for (int it = 0; it < ktiles; ++it) {
        const int buf = it & 1;
        const int kt0 = it * 64;
        storeV(buf);
        __syncthreads();
        if (it + 1 < ktiles) loadV((it + 1) * 64);
        #pragma unroll
        for (int kk = 0; kk < 64; kk += 4) {
            const int ka = kk + 2 * half;
            v2f a = ld2(&P[wqm * 16 + ml][kt0 + ka]);
            v2f b = ld2(&KV[buf][wqn * 16 + ml][ka]);
            acc = __builtin_amdgcn_wmma_f32_16x16x4_f32(
                    false, a, false, b, (short)0, acc, false, false);
        }
    }

    const int d = wqn * 16 + ml;
    const long long cbase = ((long long)bb * T_SZ + q0) * D_SZ + hh * HD_SZ + d;
    #pragma unroll
    for (int r = 0; r < 8; ++r) {
        const int rowl = wqm * 16 + 8 * half + r;
        Ctx[cbase + (long long)rowl * D_SZ] = acc[r];
    }
    wait_async_all();                  // att stores complete
}

// ---------------------------------------------------------------------------
extern "C" void kernel_launch(void* const* d_in, const int* in_sizes, int n_in,
                              void* d_out, int out_size, void* d_ws, size_t ws_size,
                              hipStream_t stream)
{
    const float*         x    = (const float*)d_in[0];
    const unsigned char* mask = (const unsigned char*)d_in[1];
    const float* wq = (const float*)d_in[2]; const float* bq = (const float*)d_in[3];
    const float* wk = (const float*)d_in[4]; const float* bk = (const float*)d_in[5];
    const float* wv = (const float*)d_in[6]; const float* bv = (const float*)d_in[7];
    const float* wp = (const float*)d_in[8]; const float* bp = (const float*)d_in[9];

    float* out = (float*)d_out;
    float* att = out + (long long)B_SZ * T_SZ * D_SZ;   // second output, [B,H,T,T]

    const long long qkv_elems = (long long)B_SZ * T_SZ * D_SZ;
    float* Qw  = (float*)d_ws;          // [B,H,T,hd]
    float* Kw  = Qw + qkv_elems;
    float* Vw  = Kw + qkv_elems;
    float* Ctx = Vw + qkv_elems;        // [B,T,D]

    const dim3 blk(256);

    // Q/K/V projections: M=B*T=4096, N=K=D=1024, head-split writes
    const dim3 gproj(D_SZ / 64, (B_SZ * T_SZ) / 128);
    gemm_f32_wmma<1><<<gproj, blk, 0, stream>>>(x, wq, bq, Qw, B_SZ * T_SZ, D_SZ, D_SZ);
    gemm_f32_wmma<1><<<gproj, blk, 0, stream>>>(x, wk, bk, Kw, B_SZ * T_SZ, D_SZ, D_SZ);
    gemm_f32_wmma<1><<<gproj, blk, 0, stream>>>(x, wv, bv, Vw, B_SZ * T_SZ, D_SZ, D_SZ);

    // fused scores -> softmax -> att store || P@V
    const dim3 gatt(T_SZ / 32, B_SZ * H_SZ);
    attn_fused<<<gatt, blk, 0, stream>>>(Qw, Kw, Vw, mask, att, Ctx);

    // out = ctx @ wp + bp
    gemm_f32_wmma<0><<<gproj, blk, 0, stream>>>(Ctx, wp, bp, out, B_SZ * T_SZ, D_SZ, D_SZ);
}